// GNN_3j1m_hetero_70016556859578
// MI455X (gfx1250) — compile-verified
//
#include <hip/hip_runtime.h>

typedef float v2f __attribute__((ext_vector_type(2)));
typedef float v4f __attribute__((ext_vector_type(4)));
typedef float v8f __attribute__((ext_vector_type(8)));

#define WPG 4            // waves per workgroup (128 threads)
#define TILES_PER_WAVE 2 // grid-stride amortization of the weight preamble
#define LN_EPS 1e-5f

struct Params {
    const float *x, *edge_attr;
    const float *je_w1,*je_b1,*je_w2,*je_b2;
    const float *mu_w1,*mu_b1,*mu_w2,*mu_b2;
    const float *lin1_w,*lin1_b,*m1_w1,*m1_b1,*m1_w2,*m1_b2;
    const float *lin2_w,*lin2_b,*m2_w1,*m2_b1,*m2_w2,*m2_b2;
    const float *fc_w1,*fc_b1,*fc_w2,*fc_b2;
    const long long *type_id;
    float *out;
    int num_graphs, num_nodes, num_edges, num_tiles;
};

// D = A(16x4 f32) * B(4x16 f32) + C : full-precision fp32 matrix core
__device__ __forceinline__ v8f wmma_f32(v2f a, v2f b, v8f c) {
    return __builtin_amdgcn_wmma_f32_16x16x4_f32(false, a, false, b, (short)0, c,
                                                 false, false);
}

// ---- per-lane register-resident weights (A-operand layout: M=lo, K=2*hi+j) ----
// fused heterogeneous encoder: layer1 K = [je rows 0..4 ; mu rows 0..4 ; pad] (12)
//                              layer2 K = [je_w2 (16) ; mu_w2 (16)] (32)
struct EncW { v2f a1[3]; v2f a2[8]; float b1[8], b2[8]; };
struct LinW { v2f a; float b[8]; };
struct MlpW { v2f w1[4][4]; float b1[4][8]; v2f w2[4][4]; float b2[8]; };

__device__ __forceinline__ EncW load_enc_fused(const Params& p, int lo, int hi,
                                               float himask, bool isjet) {
    EncW e;
    // stacked layer-1 row K: K<5 -> je_w1[K], K<10 -> mu_w1[K-5], else 0
    int k0 = 2 * hi;
    e.a1[0].x = (k0     < 5) ? p.je_w1[k0 * 16 + lo]       : p.mu_w1[(k0 - 5) * 16 + lo];
    e.a1[0].y = (k0 + 1 < 5) ? p.je_w1[(k0 + 1) * 16 + lo] : p.mu_w1[(k0 - 4) * 16 + lo];
    e.a1[1].x = (4 + k0 < 5) ? p.je_w1[(4 + k0) * 16 + lo] : p.mu_w1[(k0 - 1) * 16 + lo];
    e.a1[1].y = p.mu_w1[k0 * 16 + lo];                       // K = 5+k0
    e.a1[2].x = p.mu_w1[3 * 16 + lo] * himask;               // K = 8 (hi=0 only)
    e.a1[2].y = p.mu_w1[4 * 16 + lo] * himask;               // K = 9 (hi=0 only)
#pragma unroll
    for (int c = 0; c < 4; ++c) {
        int k = 4 * c + 2 * hi;
        e.a2[c].x     = p.je_w2[k * 16 + lo];
        e.a2[c].y     = p.je_w2[(k + 1) * 16 + lo];
        e.a2[c + 4].x = p.mu_w2[k * 16 + lo];
        e.a2[c + 4].y = p.mu_w2[(k + 1) * 16 + lo];
    }
#pragma unroll
    for (int v = 0; v < 8; ++v) {
        e.b1[v] = isjet ? p.je_b1[v + 8 * hi] : p.mu_b1[v + 8 * hi];
        e.b2[v] = isjet ? p.je_b2[v + 8 * hi] : p.mu_b2[v + 8 * hi];
    }
    return e;
}

__device__ __forceinline__ LinW load_lin(const float* w, const float* b, int lo, int hi) {
    LinW l;
    l.a.x = w[(2 * hi) * 16 + lo];
    l.a.y = w[(2 * hi + 1) * 16 + lo];
#pragma unroll
    for (int v = 0; v < 8; ++v) l.b[v] = b[v + 8 * hi];
    return l;
}

__device__ __forceinline__ MlpW load_mlp(const float* w1, const float* b1,
                                         const float* w2, const float* b2,
                                         int lo, int hi) {
    MlpW m;
#pragma unroll
    for (int mi = 0; mi < 4; ++mi) {
#pragma unroll
        for (int c = 0; c < 4; ++c) {
            int k = 4 * c + 2 * hi;
            m.w1[mi][c].x = w1[k * 64 + 16 * mi + lo];
            m.w1[mi][c].y = w1[(k + 1) * 64 + 16 * mi + lo];
            int k2 = 16 * mi + 4 * c + 2 * hi;
            m.w2[mi][c].x = w2[k2 * 16 + lo];
            m.w2[mi][c].y = w2[(k2 + 1) * 16 + lo];
        }
#pragma unroll
        for (int v = 0; v < 8; ++v) m.b1[mi][v] = b1[16 * mi + v + 8 * hi];
    }
#pragma unroll
    for (int v = 0; v < 8; ++v) m.b2[v] = b2[v + 8 * hi];
    return m;
}

// Relayout D-accumulator (N=lo, M=v+8*hi) into B K-chunks (K=4c+2*hi+j, N=lo)
__device__ __forceinline__ void d2b(const float d[8], v2f b[4], int hi) {
    float s[8];
#pragma unroll
    for (int v = 0; v < 8; ++v) s[v] = __shfl_xor(d[v], 16, 32);
    v2f t;
    t.x = hi ? s[2] : d[0]; t.y = hi ? s[3] : d[1]; b[0] = t;
    t.x = hi ? s[6] : d[4]; t.y = hi ? s[7] : d[5]; b[1] = t;
    t.x = hi ? d[2] : s[0]; t.y = hi ? d[3] : s[1]; b[2] = t;
    t.x = hi ? d[6] : s[4]; t.y = hi ? d[7] : s[5]; b[3] = t;
}

// per-node LayerNorm over 16 features
__device__ __forceinline__ void ln16(float h[8], int relu) {
    float s = 0.f, ss = 0.f;
#pragma unroll
    for (int v = 0; v < 8; ++v) { s += h[v]; ss += h[v] * h[v]; }
    s  += __shfl_xor(s, 16, 32);
    ss += __shfl_xor(ss, 16, 32);
    float m   = s * 0.0625f;
    float var = ss * 0.0625f - m * m;
    float r   = rsqrtf(var + LN_EPS);
#pragma unroll
    for (int v = 0; v < 8; ++v) {
        float y = (h[v] - m) * r;
        h[v] = relu ? fmaxf(y, 0.f) : y;
    }
}

// fused heterogeneous encoder: h = MLP_type(x), both types in one WMMA chain
__device__ __forceinline__ void encoder_fused(const EncW& E,
                                              float x0, float x1, float x2,
                                              float x3, float x4,
                                              float jm, float mm, int hi, float h[8]) {
    // stacked-K B chunks (column mask jm/mm selects the encoder per node)
    v2f B0, B1, B2;
    B0.x = (hi ? x2 : x0) * jm;
    B0.y = (hi ? x3 : x1) * jm;
    B1.x = hi ? x1 * mm : x4 * jm;     // K=4+2hi
    B1.y = hi ? x2 * mm : x0 * mm;     // K=5+2hi
    B2.x = hi ? 0.f : x3 * mm;         // K=8
    B2.y = hi ? 0.f : x4 * mm;         // K=9
    v8f a = {};
    a = wmma_f32(E.a1[0], B0, a);
    a = wmma_f32(E.a1[1], B1, a);
    a = wmma_f32(E.a1[2], B2, a);
    float t[8];
#pragma unroll
    for (int v = 0; v < 8; ++v) t[v] = fmaxf((float)a[v] + E.b1[v], 0.f);
    v2f Bt[4]; d2b(t, Bt, hi);
    v8f a2 = {};
#pragma unroll
    for (int c = 0; c < 8; ++c) {      // K=32: [t*jm ; t*mm]
        float sc = (c < 4) ? jm : mm;
        v2f b; b.x = Bt[c & 3].x * sc; b.y = Bt[c & 3].y * sc;
        a2 = wmma_f32(E.a2[c], b, a2);
    }
#pragma unroll
    for (int v = 0; v < 8; ++v) h[v] = (float)a2[v] + E.b2[v];
}

// GINEConv: agg[t] = sum_{s!=t} relu(h[s] + Lin(ea)); h = MLP16->64->16(h + agg)
__device__ __forceinline__ void gine(float* smsg, float h[8], const v2f Bea[3],
                                     const LinW& L, const MlpW& M,
                                     int lo, int hi, const int* src_l, const int* in_e) {
#pragma unroll
    for (int q = 0; q < 3; ++q) {
        v8f el = {};
        el = wmma_f32(L.a, Bea[q], el);          // elin^T for 16 edges
        int sidx = src_l[q] + (hi << 4);         // cross-lane h[src] gather
        float mg[8];
#pragma unroll
        for (int v = 0; v < 8; ++v)
            mg[v] = fmaxf(__shfl(h[v], sidx, 32) + (float)el[v] + L.b[v], 0.f);
        v4f* mp = (v4f*)(smsg + (q * 16 + lo) * 16 + 8 * hi);   // 16B-aligned
        v4f m0, m1;
        m0.x = mg[0]; m0.y = mg[1]; m0.z = mg[2]; m0.w = mg[3];
        m1.x = mg[4]; m1.y = mg[5]; m1.z = mg[6]; m1.w = mg[7];
        mp[0] = m0; mp[1] = m1;
    }
    // deterministic target-side aggregation (3 static incoming edges per node)
    float u[8];
#pragma unroll
    for (int v = 0; v < 8; ++v) u[v] = h[v];
#pragma unroll
    for (int r = 0; r < 3; ++r) {
        const v4f* ap = (const v4f*)(smsg + in_e[r] * 16 + 8 * hi);
        v4f a0 = ap[0], a1 = ap[1];
        u[0] += a0.x; u[1] += a0.y; u[2] += a0.z; u[3] += a0.w;
        u[4] += a1.x; u[5] += a1.y; u[6] += a1.z; u[7] += a1.w;
    }
    // MLP 16 -> 64 -> 16 (all in matrix pipe)
    v2f Bu[4]; d2b(u, Bu, hi);
    float z[4][8];
#pragma unroll
    for (int m = 0; m < 4; ++m) {
        v8f acc = {};
#pragma unroll
        for (int c = 0; c < 4; ++c) acc = wmma_f32(M.w1[m][c], Bu[c], acc);
#pragma unroll
        for (int v = 0; v < 8; ++v) z[m][v] = fmaxf((float)acc[v] + M.b1[m][v], 0.f);
    }
    v8f acc2 = {};
#pragma unroll
    for (int m = 0; m < 4; ++m) {
        v2f Bz[4]; d2b(z[m], Bz, hi);
#pragma unroll
        for (int c = 0; c < 4; ++c) acc2 = wmma_f32(M.w2[m][c], Bz[c], acc2);
    }
#pragma unroll
    for (int v = 0; v < 8; ++v) h[v] = (float)acc2[v] + M.b2[v];
}

__global__ __launch_bounds__(WPG * 32, 1)
void gnn_3j1m_kernel(Params p) {
    __shared__ float smsg_all[WPG * 768];   // 48 edges x 16 feat per wave
    __shared__ float sgf_all[WPG * 512];    // 32 feat x 16 graphs per wave

    const int lane = threadIdx.x & 31;
    const int wave = threadIdx.x >> 5;
    const int lo   = lane & 15;
    const int hi   = lane >> 4;
    const int wave_id = blockIdx.x * WPG + wave;
    const int nwaves  = gridDim.x * WPG;
    if (wave_id >= p.num_tiles) return;         // no barriers anywhere: safe

    float* smsg = smsg_all + wave * 768;
    float* sgf  = sgf_all + wave * 512;
    const float himask = hi ? 0.f : 1.f;

    // node type has period 4 and every tile base is % 16 == 0 -> invariant per lane
    const bool isjet = (p.type_id[lo] == 0);
    const float jm = isjet ? 1.f : 0.f;
    const float mm = 1.f - jm;

    // one-time per-wave weight load straight into VGPRs (L2-resident, ~30KB total)
    EncW Ef  = load_enc_fused(p, lo, hi, himask, isjet);
    LinW L1  = load_lin(p.lin1_w, p.lin1_b, lo, hi);
    LinW L2  = load_lin(p.lin2_w, p.lin2_b, lo, hi);
    MlpW M1  = load_mlp(p.m1_w1, p.m1_b1, p.m1_w2, p.m1_b2, lo, hi);
    MlpW M2  = load_mlp(p.m2_w1, p.m2_b1, p.m2_w2, p.m2_b2, lo, hi);

    // static 4-node fully-connected topology (edge k = 3*i + j - (j>i))
    int src_l[3];
#pragma unroll
    for (int q = 0; q < 3; ++q) {
        int e48 = q * 16 + lo;
        int g4  = e48 / 12;
        int k   = e48 - 12 * g4;
        src_l[q] = 4 * g4 + (k / 3);
    }
    int in_e[3];
    {
        int t_l = lo & 3, g4 = lo >> 2;
#pragma unroll
        for (int r = 0; r < 3; ++r) {
            int ir = r + ((r >= t_l) ? 1 : 0);
            int kr = 3 * ir + t_l - ((t_l > ir) ? 1 : 0);
            in_e[r] = 12 * g4 + kr;
        }
    }

    // grid-stride over tiles: amortize the weight/topology preamble
    for (int tile = wave_id; tile < p.num_tiles; tile += nwaves) {

        for (int nt = 0; nt < 4; ++nt) {
            const int node_base = tile * 64 + nt * 16;
            int ng = node_base + lo;
            if (ng >= p.num_nodes) ng = p.num_nodes - 1;   // clamp; pads never stored
            const float* xp = p.x + ng * 5;
            float x0 = xp[0], x1 = xp[1], x2 = xp[2], x3 = xp[3], x4 = xp[4];

            float h[8];
            encoder_fused(Ef, x0, x1, x2, x3, x4, jm, mm, hi, h);

            // edge_attr B-operands for this node-tile (K=4 exactly, b64 loads)
            v2f Bea[3];
            const int ebase = tile * 192 + nt * 48;
#pragma unroll
            for (int q = 0; q < 3; ++q) {
                int eg = ebase + q * 16 + lo;
                if (eg >= p.num_edges) eg = p.num_edges - 1;
                const v2f* e2 = (const v2f*)(p.edge_attr + eg * 4);
                Bea[q] = e2[hi];
            }
            if (nt < 3) {   // prefetch next node-tile's streams (global_prefetch_b8)
                __builtin_prefetch(p.x + (size_t)(node_base + 16 + lo) * 5, 0, 0);
                __builtin_prefetch(p.edge_attr + (size_t)(ebase + 48 + lo) * 4, 0, 0);
            }

            gine(smsg, h, Bea, L1, M1, lo, hi, src_l, in_e);
            ln16(h, 1);
            gine(smsg, h, Bea, L2, M2, lo, hi, src_l, in_e);
            ln16(h, 1);

            // per-graph mean/max pooling over the 4 nodes (aligned lane groups of 4)
            float pm[8], px[8];
#pragma unroll
            for (int v = 0; v < 8; ++v) {
                float a1 = __shfl_xor(h[v], 1, 32);
                float t  = h[v] + a1;
                t += __shfl_xor(t, 2, 32);
                pm[v] = t * 0.25f;
                float m = fmaxf(h[v], a1);
                m = fmaxf(m, __shfl_xor(m, 2, 32));
                px[v] = m;
            }
            // graph-level LayerNorm over the 32 pooled features
            float s = 0.f, ss = 0.f;
#pragma unroll
            for (int v = 0; v < 8; ++v) {
                s  += pm[v] + px[v];
                ss += pm[v] * pm[v] + px[v] * px[v];
            }
            s  += __shfl_xor(s, 16, 32);
            ss += __shfl_xor(ss, 16, 32);
            float gmn = s * (1.f / 32.f);
            float gvr = ss * (1.f / 32.f) - gmn * gmn;
            float gr  = rsqrtf(gvr + LN_EPS);
            const int gcol = 4 * nt + (lo >> 2);
            const int r4   = lo & 3;
#pragma unroll
            for (int v = 0; v < 8; ++v) {
                if ((v & 3) == r4) {   // spread replicated writes across the 4 lanes
                    sgf[(v + 8 * hi) * 16 + gcol]      = (pm[v] - gmn) * gr;
                    sgf[(16 + v + 8 * hi) * 16 + gcol] = (px[v] - gmn) * gr;
                }
            }
        }

        // final FC: g[16 graphs x 32] -> 64 -> 1, graphs on lanes
        v2f Bg[8];
#pragma unroll
        for (int c = 0; c < 8; ++c) {
            int k = 4 * c + 2 * hi;
            Bg[c].x = sgf[k * 16 + lo];
            Bg[c].y = sgf[(k + 1) * 16 + lo];
        }
        float s = 0.f;
#pragma unroll
        for (int m = 0; m < 4; ++m) {
            v8f acc = {};
#pragma unroll
            for (int c = 0; c < 8; ++c) {
                int k = 4 * c + 2 * hi;
                v2f a;
                a.x = p.fc_w1[k * 64 + 16 * m + lo];
                a.y = p.fc_w1[(k + 1) * 64 + 16 * m + lo];
                acc = wmma_f32(a, Bg[c], acc);
            }
#pragma unroll
            for (int v = 0; v < 8; ++v) {
                float zf = fmaxf((float)acc[v] + p.fc_b1[16 * m + v + 8 * hi], 0.f);
                s += zf * p.fc_w2[16 * m + 8 * hi + v];
            }
        }
        s += __shfl_xor(s, 16, 32);
        if (hi == 0) {
            int gg = tile * 16 + lo;
            if (gg < p.num_graphs) p.out[gg] = s + p.fc_b2[0];
        }
    }
}

extern "C" void kernel_launch(void* const* d_in, const int* in_sizes, int n_in,
                              void* d_out, int out_size, void* d_ws, size_t ws_size,
                              hipStream_t stream) {
    (void)n_in; (void)out_size; (void)d_ws; (void)ws_size;
    Params p;
    p.x         = (const float*)d_in[0];
    p.edge_attr = (const float*)d_in[1];
    p.je_w1 = (const float*)d_in[2];  p.je_b1 = (const float*)d_in[3];
    p.je_w2 = (const float*)d_in[4];  p.je_b2 = (const float*)d_in[5];
    p.mu_w1 = (const float*)d_in[6];  p.mu_b1 = (const float*)d_in[7];
    p.mu_w2 = (const float*)d_in[8];  p.mu_b2 = (const float*)d_in[9];
    p.lin1_w= (const float*)d_in[10]; p.lin1_b= (const float*)d_in[11];
    p.m1_w1 = (const float*)d_in[12]; p.m1_b1 = (const float*)d_in[13];
    p.m1_w2 = (const float*)d_in[14]; p.m1_b2 = (const float*)d_in[15];
    p.lin2_w= (const float*)d_in[16]; p.lin2_b= (const float*)d_in[17];
    p.m2_w1 = (const float*)d_in[18]; p.m2_b1 = (const float*)d_in[19];
    p.m2_w2 = (const float*)d_in[20]; p.m2_b2 = (const float*)d_in[21];
    p.fc_w1 = (const float*)d_in[22]; p.fc_b1 = (const float*)d_in[23];
    p.fc_w2 = (const float*)d_in[24]; p.fc_b2 = (const float*)d_in[25];
    p.type_id = (const long long*)d_in[26];
    p.out = (float*)d_out;
    const int N = in_sizes[0] / 5;
    p.num_nodes  = N;
    p.num_edges  = in_sizes[1] / 4;
    p.num_graphs = N / 4;                     // NPG = 4
    p.num_tiles  = (p.num_graphs + 15) / 16;  // 16 graphs per wave
    const int waves_needed = (p.num_tiles + TILES_PER_WAVE - 1) / TILES_PER_WAVE;
    const int blocks = (waves_needed + WPG - 1) / WPG;
    gnn_3j1m_kernel<<<blocks, WPG * 32, 0, stream>>>(p);
}